// PoseExtrinsic_48447231098899
// MI455X (gfx1250) — compile-verified
//
#include <hip/hip_runtime.h>
#include <cstdint>
#include <cstddef>

typedef float v4f __attribute__((ext_vector_type(4)));

#define BLOCK      256
#define RPT        8            // rays per thread
#define IMAGE_NUM  128

__global__ __launch_bounds__(BLOCK) void pose_ray_kernel(
    const v4f* __restrict__ xyzw,
    const int* __restrict__ ids,
    const float* __restrict__ pose,
    float* __restrict__ out,
    int n)
{
    // Top-3 rows of each 4x4 pose, one v4f per row: 128 images * 3 rows * 16B = 6 KB.
    // Row r of image i lives at byte offset (i*48 + r*16): 16B-aligned for b128 DS loads.
    __shared__ v4f sP[IMAGE_NUM * 3];

    const int t = threadIdx.x;

    // ---- Stage pose rows into LDS with CDNA5 async global->LDS b128 copies ----
    // 384 rows; round 0 covers e=0..255 (unconditional), round 1 covers e=256..383.
    // Passing &sP[e] escapes sP's address so, with the "memory" clobber, the
    // compiler must assume the asm writes sP (keeps the ds_load gathers alive).
    auto stage = [&](int e) {
        int img = e / 3;
        int r   = e - img * 3;
        unsigned lds_off = (unsigned)(e * 16);          // (img*3 + r) * 16 bytes
        const float* gp  = pose + (img * 16 + r * 4);   // row r of pose[img]
        v4f* lp = &sP[e];                               // escapes sP's address
        asm volatile("global_load_async_to_lds_b128 %[ldsoff], %[gaddr], off"
                     :
                     : [ldsoff] "v"(lds_off), [gaddr] "v"(gp), "v"(lp)
                     : "memory");
    };
    stage(t);                                           // e = t, always < 384
    if (t < IMAGE_NUM * 3 - BLOCK) stage(t + BLOCK);    // e = t+256 < 384
    asm volatile("s_wait_asynccnt 0" ::: "memory");
    __syncthreads();

    float* o_out = out;
    float* d_dir = out + (size_t)3 * (size_t)n;

    const int base = blockIdx.x * (BLOCK * RPT) + t;

    auto body = [&](int i) {
        // Streaming inputs: nontemporal b128 / b32 loads
        v4f x  = __builtin_nontemporal_load(xyzw + i);
        int id = __builtin_nontemporal_load(ids + i);
        __builtin_assume(id >= 0 && id < IMAGE_NUM);

        // Pose gather straight from shared memory -> ds_load_b128 x3
        v4f r0 = sP[id * 3 + 0];
        v4f r1 = sP[id * 3 + 1];
        v4f r2 = sP[id * 3 + 2];

        // d_r = P[r][0]x + P[r][1]y + P[r][2]z + P[r][3]*(w-1),  o_r = P[r][3]
        float w1 = x.w - 1.0f;
        float dx = fmaf(r0.x, x.x, fmaf(r0.y, x.y, fmaf(r0.z, x.z, r0.w * w1)));
        float dy = fmaf(r1.x, x.x, fmaf(r1.y, x.y, fmaf(r1.z, x.z, r1.w * w1)));
        float dz = fmaf(r2.x, x.x, fmaf(r2.y, x.y, fmaf(r2.z, x.z, r2.w * w1)));

        float inv = __builtin_amdgcn_rsqf(fmaf(dx, dx, fmaf(dy, dy, dz * dz)));

        int oi = 3 * i;   // fits in int32; enables saddr + 32-bit voffset addressing
        __builtin_nontemporal_store(r0.w,     o_out + oi + 0);
        __builtin_nontemporal_store(r1.w,     o_out + oi + 1);
        __builtin_nontemporal_store(r2.w,     o_out + oi + 2);
        __builtin_nontemporal_store(dx * inv, d_dir + oi + 0);
        __builtin_nontemporal_store(dy * inv, d_dir + oi + 1);
        __builtin_nontemporal_store(dz * inv, d_dir + oi + 2);
    };

    if (base + (RPT - 1) * BLOCK < n) {
        // Fast path: whole chunk in range, no per-iteration exec-mask churn.
        #pragma unroll
        for (int k = 0; k < RPT; ++k)
            body(base + k * BLOCK);
    } else {
        // Rare tail: keep it rolled to minimize static code size.
        #pragma unroll 1
        for (int k = 0; k < RPT; ++k) {
            int i = base + k * BLOCK;
            if (i < n) body(i);
        }
    }
}

extern "C" void kernel_launch(void* const* d_in, const int* in_sizes, int n_in,
                              void* d_out, int out_size, void* d_ws, size_t ws_size,
                              hipStream_t stream) {
    const v4f*   xyzw = (const v4f*)d_in[0];
    const int*   ids  = (const int*)d_in[1];
    const float* pose = (const float*)d_in[2];
    float*       out  = (float*)d_out;

    int n = in_sizes[1];                      // number of rays
    int grid = (n + BLOCK * RPT - 1) / (BLOCK * RPT);
    pose_ray_kernel<<<dim3(grid), dim3(BLOCK), 0, stream>>>(xyzw, ids, pose, out, n);
}